// Phi_r_87325275062697
// MI455X (gfx1250) — compile-verified
//
#include <hip/hip_runtime.h>

typedef float v2f __attribute__((ext_vector_type(2)));
typedef float v4f __attribute__((ext_vector_type(4)));
typedef float v8f __attribute__((ext_vector_type(8)));

#define HH 256
#define WW 256
#define HWSZ (256*256)
#define HP 258
#define WP 258
#define HPWP (258*258)

// ---------------- helpers ----------------

__device__ __forceinline__ float softplus10(float v) {
    float z = 10.0f * v;
    float r = fmaxf(z, 0.0f) + log1pf(expf(-fabsf(z)));
    return r * 0.1f;
}

// Raw padded x, NCHW [N*10][HP][WP] (for the PAC patch reads).
__global__ void pad_raw_kernel(const float* __restrict__ x,
                               float* __restrict__ xp, int NC) {
    int idx = blockIdx.x * blockDim.x + threadIdx.x;
    int total = NC * HPWP;
    if (idx >= total) return;
    int nc = idx / HPWP;
    int r  = idx - nc * HPWP;
    int yy = r / WP;
    int xx = r - yy * WP;
    float v = 0.0f;
    if (yy >= 1 && yy <= HH && xx >= 1 && xx <= WW)
        v = x[(size_t)nc * HWSZ + (size_t)(yy - 1) * WW + (xx - 1)];
    xp[idx] = v;
}

// relu(x) padded, NHWC [N][HP][WP][12] (channels 10,11 zero) for conv1.
__global__ void pad_relu_nhwc_kernel(const float* __restrict__ x,
                                     float* __restrict__ xp) {
    int idx = blockIdx.x * blockDim.x + threadIdx.x;
    const int total = 4 * HPWP * 12;
    if (idx >= total) return;
    int ci  = idx % 12;
    int rem = idx / 12;
    int xx  = rem % WP;
    rem /= WP;
    int yy  = rem % HP;
    int n   = rem / HP;
    float v = 0.0f;
    if (ci < 10 && yy >= 1 && yy <= HH && xx >= 1 && xx <= WW)
        v = fmaxf(x[((size_t)(n * 10 + ci)) * HWSZ +
                    (size_t)(yy - 1) * WW + (xx - 1)], 0.0f);
    xp[idx] = v;
}

// Repack weights [M][CIN][3][3] -> [Mpad][9*CINP], k = tap*CINP + ci, zero pad.
__global__ void repack_w_kernel(const float* __restrict__ src,
                                float* __restrict__ dst,
                                int M, int CIN, int CINP, int Mpad) {
    int KPAD = 9 * CINP;
    int idx = blockIdx.x * blockDim.x + threadIdx.x;
    int total = Mpad * KPAD;
    if (idx >= total) return;
    int m = idx / KPAD;
    int r = idx - m * KPAD;
    int tap = r / CINP;
    int ci  = r - tap * CINP;
    float v = 0.0f;
    if (m < M && ci < CIN)
        v = src[((size_t)m * CIN + ci) * 9 + tap];
    dst[idx] = v;
}

// ---------------- NHWC implicit-GEMM 3x3 conv, WMMA fp32 ----------------
// act : [N][HP][WP][CINP] zero halo / zero channel pad
// wpad: [Mpad][9*CINP], k = tap*CINP + ci
// out : PADOUT ? [N][HP][WP][COUT] (interior) : [N][H][W][COUT]
// Wave computes 32 output channels x 64 pixels of one row:
//   per K-block: 2 A b64 + 4 B b64 loads (all immediate offset) + 8 WMMA.
template <int CINP, int COUT, bool STORE_RELU, bool PADOUT, bool VECSTORE>
__global__ __launch_bounds__(128)
void conv3x3_wmma_nhwc(const float* __restrict__ act,
                       const float* __restrict__ wpad,
                       float* __restrict__ out,
                       int coutReal) {
    constexpr int KPAD = 9 * CINP;

    const int lane = threadIdx.x & 31;
    const int half = lane >> 4;   // selects K pair (ISA A/B half-wave layout)
    const int l16  = lane & 15;
    const int wave = threadIdx.x >> 5;

    const int y  = blockIdx.x;        // output row
    const int m0 = blockIdx.y * 32;   // 2 Cout tiles per wave
    const int n  = blockIdx.z;
    const int x0 = wave * 64;

    // pixel (y, x0+l16) in padded coords reads taps at (y+dy, x+dx)
    const float* base0 =
        act + ((size_t)(n * HP + y) * WP + (x0 + l16)) * CINP + 2 * half;
    const float* wb0 = wpad + (size_t)(m0 + l16) * KPAD + 2 * half;
    const float* wb1 = wb0 + (size_t)16 * KPAD;

    v8f cacc[2][4] = {};

    for (int dy = 0; dy < 3; ++dy) {
        const float* bdy = base0 + (size_t)dy * WP * CINP;
        const float* wdy0 = wb0 + dy * 3 * CINP;
        const float* wdy1 = wb1 + dy * 3 * CINP;
#pragma unroll
        for (int dx = 0; dx < 3; ++dx) {
#pragma unroll
            for (int ci = 0; ci < CINP; ci += 4) {
                const int ko = dx * CINP + ci;   // compile-time immediates
                v2f a0 = *(const v2f*)(wdy0 + ko);
                v2f a1 = *(const v2f*)(wdy1 + ko);
                v2f b0 = *(const v2f*)(bdy + ko);
                v2f b1 = *(const v2f*)(bdy + ko + 16 * CINP);
                v2f b2 = *(const v2f*)(bdy + ko + 32 * CINP);
                v2f b3 = *(const v2f*)(bdy + ko + 48 * CINP);
                cacc[0][0] = __builtin_amdgcn_wmma_f32_16x16x4_f32(
                    false, a0, false, b0, (short)0, cacc[0][0], false, false);
                cacc[0][1] = __builtin_amdgcn_wmma_f32_16x16x4_f32(
                    false, a0, false, b1, (short)0, cacc[0][1], false, false);
                cacc[0][2] = __builtin_amdgcn_wmma_f32_16x16x4_f32(
                    false, a0, false, b2, (short)0, cacc[0][2], false, false);
                cacc[0][3] = __builtin_amdgcn_wmma_f32_16x16x4_f32(
                    false, a0, false, b3, (short)0, cacc[0][3], false, false);
                cacc[1][0] = __builtin_amdgcn_wmma_f32_16x16x4_f32(
                    false, a1, false, b0, (short)0, cacc[1][0], false, false);
                cacc[1][1] = __builtin_amdgcn_wmma_f32_16x16x4_f32(
                    false, a1, false, b1, (short)0, cacc[1][1], false, false);
                cacc[1][2] = __builtin_amdgcn_wmma_f32_16x16x4_f32(
                    false, a1, false, b2, (short)0, cacc[1][2], false, false);
                cacc[1][3] = __builtin_amdgcn_wmma_f32_16x16x4_f32(
                    false, a1, false, b3, (short)0, cacc[1][3], false, false);
            }
        }
    }

    // D VGPR r -> (M = r + 8*half, N = l16); channels are contiguous in NHWC.
#pragma unroll
    for (int mt = 0; mt < 2; ++mt) {
#pragma unroll
        for (int s = 0; s < 4; ++s) {
            int x = x0 + s * 16 + l16;
            size_t pix = PADOUT ? ((size_t)(n * HP + (y + 1)) * WP + (x + 1))
                                : ((size_t)(n * HH + y) * WW + x);
            int cobase = m0 + mt * 16 + 8 * half;
            float* o = out + pix * COUT + cobase;
            v8f cc = cacc[mt][s];
            if (STORE_RELU) {
#pragma unroll
                for (int r = 0; r < 8; ++r) cc[r] = fmaxf(cc[r], 0.0f);
            }
            if (VECSTORE) {
                *(v4f*)(o + 0) = (v4f){cc[0], cc[1], cc[2], cc[3]};
                *(v4f*)(o + 4) = (v4f){cc[4], cc[5], cc[6], cc[7]};
            } else {
#pragma unroll
                for (int r = 0; r < 8; ++r)
                    if (cobase + r < coutReal) o[r] = cc[r];
            }
        }
    }
}

// ---------------- fused guide / KMAT / PAC epilogue ----------------
// params NHWC [4][256][256][60], xpad NCHW [4][10][HP][WP], out [4][10][256][256]
__global__ void pac_final_kernel(const float* __restrict__ params,
                                 const float* __restrict__ xpad,
                                 const float* __restrict__ pacw,
                                 const float* __restrict__ pacb,
                                 float* __restrict__ out) {
    int idx = blockIdx.x * blockDim.x + threadIdx.x;
    const int total = 4 * 10 * HWSZ;
    if (idx >= total) return;
    int px = idx % HWSZ;
    int nj = idx / HWSZ;
    int j  = nj % 10;
    int n  = nj / 10;
    if (j == 9) { out[idx] = 0.0f; return; }
    int y = px / WW;
    int x = px - y * WW;
    int tp = j + 1;

    const float* pb = params + ((size_t)n * HWSZ + (size_t)y * WW + x) * 60;
    float kraw = pb[ 0 + tp];
    float m1d  = pb[10 + tp];
    float m2d  = pb[20 + tp];
    float graw = pb[30 + tp];
    float vx   = pb[40 + tp];
    float vy   = pb[50 + tp];

    float kappa = softplus10(kraw);
    float gamma = softplus10(graw);
    float H11 = gamma + vx * vx;
    float H22 = gamma + vy * vy;
    float H12 = vx * vy;
    float i11 = 1.0f / H11;
    float i22 = 1.0f / H22;

    float wk[9];
    wk[0] = -0.5f * H12;
    wk[1] = -i22 + m1d;
    wk[2] =  0.5f * H12;
    wk[3] = -i11 - m2d;
    wk[4] = kappa + 2.0f * H11 + 2.0f * H22 + 1.0f;
    wk[5] = -i11 + m2d;
    wk[6] =  0.5f * H12;
    wk[7] = -i22 - m1d;
    wk[8] = -0.5f * H12;

    const float* xb = xpad + (size_t)(n * 10 + tp) * HPWP + (size_t)y * WP + x;
    float acc = 0.0f;
#pragma unroll
    for (int d = 0; d < 3; ++d)
#pragma unroll
        for (int e = 0; e < 3; ++e)
            acc += pacw[d * 3 + e] * wk[d * 3 + e] * xb[(size_t)d * WP + e];

    out[idx] = acc + pacb[0];
}

// ---------------- launcher ----------------

extern "C" void kernel_launch(void* const* d_in, const int* in_sizes, int n_in,
                              void* d_out, int out_size, void* d_ws, size_t ws_size,
                              hipStream_t stream) {
    const float* x    = (const float*)d_in[0];
    const float* w1   = (const float*)d_in[1];  // [32][10][3][3]
    const float* w2   = (const float*)d_in[2];  // [64][32][3][3]
    const float* w3   = (const float*)d_in[3];  // [60][64][3][3]
    const float* pacw = (const float*)d_in[4];
    const float* pacb = (const float*)d_in[5];
    float* out = (float*)d_out;

    float* ws = (float*)d_ws;
    size_t off = 0;
    // raw padded x (NCHW) for PAC
    float* xpad = ws + off; off += (size_t)4 * 10 * HPWP;          // 10.65 MB
    // p2 region (NHWC C=64); xpadr (NHWC C=12, 12.8MB) aliases its start:
    // xpadr is dead once conv1 finishes, before p2 is zeroed/written.
    float* p2   = ws + off;
    float* xpadr = p2; off += (size_t)4 * HPWP * 64;               // 68.2 MB
    // p1 (NHWC C=32, 34.1MB) aliases params (NHWC C=60, 62.9MB): p1 dead
    // before conv3 writes params.
    float* p1  = ws + off;
    float* prm = ws + off; off += (size_t)4 * HWSZ * 60;           // 62.9 MB
    float* wp1 = ws + off; off += 32 * 9 * 12;
    float* wp2 = ws + off; off += 64 * 9 * 32;
    float* wp3 = ws + off; off += 64 * 9 * 64;
    (void)ws_size; (void)in_sizes; (void)n_in; (void)out_size;

    // zero p1 halos (whole buffer)
    hipMemsetAsync(p1, 0, (size_t)4 * HPWP * 32 * sizeof(float), stream);

    {
        int total = 4 * 10 * HPWP;
        pad_raw_kernel<<<(total + 255) / 256, 256, 0, stream>>>(x, xpad, 40);
        int total2 = 4 * HPWP * 12;
        pad_relu_nhwc_kernel<<<(total2 + 255) / 256, 256, 0, stream>>>(x, xpadr);
    }
    {
        int t1 = 32 * 9 * 12;
        repack_w_kernel<<<(t1 + 255) / 256, 256, 0, stream>>>(w1, wp1, 32, 10, 12, 32);
        int t2 = 64 * 9 * 32;
        repack_w_kernel<<<(t2 + 255) / 256, 256, 0, stream>>>(w2, wp2, 64, 32, 32, 64);
        int t3 = 64 * 9 * 64;
        repack_w_kernel<<<(t3 + 255) / 256, 256, 0, stream>>>(w3, wp3, 60, 64, 64, 64);
    }

    dim3 blk(128, 1, 1);
    // conv1: relu(x)*w1 -> relu stored into p1 (NHWC padded), Cout=32
    conv3x3_wmma_nhwc<12, 32, true, true, true>
        <<<dim3(256, 1, 4), blk, 0, stream>>>(xpadr, wp1, p1, 32);

    // zero p2 halos after xpadr (aliased) is consumed
    hipMemsetAsync(p2, 0, (size_t)4 * HPWP * 64 * sizeof(float), stream);

    // conv2: p1 * w2 -> p2 (NHWC padded), Cout=64
    conv3x3_wmma_nhwc<32, 64, false, true, true>
        <<<dim3(256, 2, 4), blk, 0, stream>>>(p1, wp2, p2, 64);

    // conv3: p2 * w3 -> params (NHWC dense), Cout=60 (scalar masked stores)
    conv3x3_wmma_nhwc<64, 60, false, false, false>
        <<<dim3(256, 2, 4), blk, 0, stream>>>(p2, wp3, prm, 60);

    {
        int total = 4 * 10 * HWSZ;
        pac_final_kernel<<<(total + 255) / 256, 256, 0, stream>>>(
            prm, xpad, pacw, pacb, out);
    }
}